// SingleAgentGNN_22694607192564
// MI455X (gfx1250) — compile-verified
//
#include <hip/hip_runtime.h>
#include <hip/hip_bf16.h>
#include <math.h>

// ---------------------------------------------------------------------------
// SingleAgentGNN forward for MI455X (gfx1250, wave32).
// Dense GEMMs use V_WMMA_F32_16X16X4_F32 (full fp32 matrix engine path).
// Edge softmax/aggregation uses float atomics resolved in the 192MB L2.
// ---------------------------------------------------------------------------

#define HID 128
#define HEADS 4
#define CH 32
#define F_IN 10
#define LAYERS 4

typedef float v2f __attribute__((ext_vector_type(2)));
typedef float v8f __attribute__((ext_vector_type(8)));

// ---------------------------------------------------------------------------
// WMMA f32 GEMM:  D[M,Nc] = act(A[M,K] @ B[K,Nc] + bias)
// One wave computes one 16x16 tile; 8 waves (256 threads) per block.
// M, Nc multiples of 16; (M/16)*(Nc/16) multiple of 8; K multiple of 4.
// ---------------------------------------------------------------------------
__global__ __launch_bounds__(256) void gemm_wmma_f32(
    const float* __restrict__ A, const float* __restrict__ B,
    const float* __restrict__ bias, float* __restrict__ D,
    int M, int K, int Ncols, int relu) {
  const int wave = threadIdx.x >> 5;
  const int lane = threadIdx.x & 31;
  const int ntiles = Ncols >> 4;
  const int tile = blockIdx.x * 8 + wave;
  const int mt = tile / ntiles;
  const int nt = tile - mt * ntiles;
  const int half = lane >> 4;      // 0: lanes 0-15, 1: lanes 16-31
  const int r = lane & 15;
  const int row = (mt << 4) + r;   // A row for this lane
  const int col = (nt << 4) + r;   // B/C column for this lane

  v8f c = {};
  const float* __restrict__ Arow = A + (size_t)row * K;
  for (int k0 = 0; k0 < K; k0 += 4) {
    // A 16x4 f32 fragment: VGPR0 = K=k0+2*half, VGPR1 = K=k0+2*half+1
    v2f a, b;
    a.x = Arow[k0 + 2 * half];
    a.y = Arow[k0 + 2 * half + 1];
    // B 4x16 f32 fragment (row-major B[K,Nc]): same K striping across halves
    b.x = B[(size_t)(k0 + 2 * half) * Ncols + col];
    b.y = B[(size_t)(k0 + 2 * half + 1) * Ncols + col];
    c = __builtin_amdgcn_wmma_f32_16x16x4_f32(
        /*neg_a=*/false, a, /*neg_b=*/false, b,
        /*c_mod=*/(short)0, c, /*reuse_a=*/false, /*reuse_b=*/false);
  }
  // C/D layout: VGPR v -> row M = v + 8*half, col N = lane&15
#pragma unroll
  for (int v = 0; v < 8; ++v) {
    const int orow = (mt << 4) + v + 8 * half;
    float val = c[v];
    if (bias) val += bias[col];
    if (relu) val = fmaxf(val, 0.0f);
    D[(size_t)orow * Ncols + col] = val;
  }
}

// ---------------------------------------------------------------------------
// Block (128 threads = 1 node) LayerNorm helper pieces done inline per kernel.
// ---------------------------------------------------------------------------

// x0 = relu(LN(nf @ W_emb + b_emb))
__global__ __launch_bounds__(128) void embed_ln_kernel(
    const float* __restrict__ nf, const float* __restrict__ W,
    const float* __restrict__ bvec, const float* __restrict__ g,
    const float* __restrict__ bln, float* __restrict__ X) {
  const int n = blockIdx.x;
  const int ccol = threadIdx.x;
  float acc = bvec[ccol];
  const float* nfr = nf + (size_t)n * F_IN;
#pragma unroll
  for (int f = 0; f < F_IN; ++f) acc += nfr[f] * W[f * HID + ccol];

  __shared__ float red[HID];
  __shared__ float s_mu, s_rstd;
  red[ccol] = acc;
  __syncthreads();
  for (int s = 64; s > 0; s >>= 1) {
    if (ccol < s) red[ccol] += red[ccol + s];
    __syncthreads();
  }
  if (ccol == 0) s_mu = red[0] * (1.0f / HID);
  __syncthreads();
  float d = acc - s_mu;
  red[ccol] = d * d;
  __syncthreads();
  for (int s = 64; s > 0; s >>= 1) {
    if (ccol < s) red[ccol] += red[ccol + s];
    __syncthreads();
  }
  if (ccol == 0) s_rstd = rsqrtf(red[0] * (1.0f / HID) + 1e-5f);
  __syncthreads();
  float y = d * s_rstd * g[ccol] + bln[ccol];
  X[(size_t)n * HID + ccol] = fmaxf(y, 0.0f);
}

// a_s[n,h] = dot(h[n,h,:], att_s[h,:]);  a_d likewise. One wave per head.
__global__ __launch_bounds__(128) void attn_dots_kernel(
    const float* __restrict__ Hbuf, const float* __restrict__ atts,
    const float* __restrict__ attd, float* __restrict__ AS,
    float* __restrict__ AD) {
  const int n = blockIdx.x;
  const int t = threadIdx.x;
  const int head = t >> 5;
  const int lane = t & 31;
  const float v = Hbuf[(size_t)n * HID + t];
  float s = v * atts[head * CH + lane];
  float d = v * attd[head * CH + lane];
#pragma unroll
  for (int o = 16; o > 0; o >>= 1) {
    s += __shfl_down(s, o, 32);
    d += __shfl_down(d, o, 32);
  }
  if (lane == 0) {
    AS[n * HEADS + head] = s;
    AD[n * HEADS + head] = d;
  }
}

__device__ __forceinline__ float lrelu02(float x) {
  return x > 0.0f ? x : 0.2f * x;
}

// MX[n,h] = self-loop alpha (initializes the segment max)
__global__ void init_max_kernel(const float* __restrict__ AS,
                                const float* __restrict__ AD,
                                float* __restrict__ MX, int NH) {
  int i = blockIdx.x * blockDim.x + threadIdx.x;
  if (i >= NH) return;
  MX[i] = lrelu02(AS[i] + AD[i]);
}

__global__ void edge_max_kernel(const int* __restrict__ src,
                                const int* __restrict__ dst,
                                const float* __restrict__ AS,
                                const float* __restrict__ AD,
                                float* __restrict__ MX, int EH) {
  int i = blockIdx.x * blockDim.x + threadIdx.x;
  if (i >= EH) return;
  int e = i >> 2, h = i & 3;
  int s = src[e], d = dst[e];
  float a = lrelu02(AS[s * HEADS + h] + AD[d * HEADS + h]);
  atomicMax(&MX[d * HEADS + h], a);
}

// DEN[n,h] = exp(self_alpha - max)  (initializes the segment sum)
__global__ void init_den_kernel(const float* __restrict__ AS,
                                const float* __restrict__ AD,
                                const float* __restrict__ MX,
                                float* __restrict__ DEN, int NH) {
  int i = blockIdx.x * blockDim.x + threadIdx.x;
  if (i >= NH) return;
  DEN[i] = expf(lrelu02(AS[i] + AD[i]) - MX[i]);
}

__global__ void edge_expsum_kernel(const int* __restrict__ src,
                                   const int* __restrict__ dst,
                                   const float* __restrict__ AS,
                                   const float* __restrict__ AD,
                                   const float* __restrict__ MX,
                                   float* __restrict__ DEN,
                                   float* __restrict__ WEDGE, int EH) {
  int i = blockIdx.x * blockDim.x + threadIdx.x;
  if (i >= EH) return;
  int e = i >> 2, h = i & 3;
  int s = src[e], d = dst[e];
  float a = lrelu02(AS[s * HEADS + h] + AD[d * HEADS + h]);
  float w = expf(a - MX[d * HEADS + h]);
  WEDGE[i] = w;
  atomicAdd(&DEN[d * HEADS + h], w);
}

// OUT[n,:] = coeff_self * h[n,:]   (self-loop contribution seeds the output)
__global__ __launch_bounds__(128) void agg_init_kernel(
    const float* __restrict__ Hbuf, const float* __restrict__ AS,
    const float* __restrict__ AD, const float* __restrict__ MX,
    const float* __restrict__ DEN, float* __restrict__ OUT) {
  const int n = blockIdx.x;
  const int t = threadIdx.x;
  const int head = t >> 5;
  const int ih = n * HEADS + head;
  float wself = expf(lrelu02(AS[ih] + AD[ih]) - MX[ih]);
  float coeff = wself / (DEN[ih] + 1e-16f);
  OUT[(size_t)n * HID + t] = coeff * Hbuf[(size_t)n * HID + t];
}

// OUT[dst,:] += (w_e / denom[dst]) * h[src,:].  128 threads per edge, 2/block.
__global__ __launch_bounds__(256) void edge_agg_kernel(
    const int* __restrict__ src, const int* __restrict__ dst,
    const float* __restrict__ Hbuf, const float* __restrict__ WEDGE,
    const float* __restrict__ DEN, float* __restrict__ OUT, int E) {
  const int t = threadIdx.x & 127;
  const int el = threadIdx.x >> 7;
  const int e = blockIdx.x * 2 + el;
  if (e >= E) return;
  const int s = src[e], d = dst[e];
  const int head = t >> 5;
  float coeff = WEDGE[e * HEADS + head] / (DEN[d * HEADS + head] + 1e-16f);
  atomicAdd(&OUT[(size_t)d * HID + t], coeff * Hbuf[(size_t)s * HID + t]);
}

// x = LN(x + relu(OUT + bg)) * g + b   (in place, one node per block)
__global__ __launch_bounds__(128) void post_ln_kernel(
    float* __restrict__ X, const float* __restrict__ OUT,
    const float* __restrict__ bg, const float* __restrict__ g,
    const float* __restrict__ bln) {
  const int n = blockIdx.x;
  const int ccol = threadIdx.x;
  float y = X[(size_t)n * HID + ccol] +
            fmaxf(OUT[(size_t)n * HID + ccol] + bg[ccol], 0.0f);

  __shared__ float red[HID];
  __shared__ float s_mu, s_rstd;
  red[ccol] = y;
  __syncthreads();
  for (int s = 64; s > 0; s >>= 1) {
    if (ccol < s) red[ccol] += red[ccol + s];
    __syncthreads();
  }
  if (ccol == 0) s_mu = red[0] * (1.0f / HID);
  __syncthreads();
  float d = y - s_mu;
  red[ccol] = d * d;
  __syncthreads();
  for (int s = 64; s > 0; s >>= 1) {
    if (ccol < s) red[ccol] += red[ccol + s];
    __syncthreads();
  }
  if (ccol == 0) s_rstd = rsqrtf(red[0] * (1.0f / HID) + 1e-5f);
  __syncthreads();
  X[(size_t)n * HID + ccol] = d * s_rstd * g[ccol] + bln[ccol];
}

// logits = H2[n,:64] @ pW3 + pb3 -> d_out[n*2 .. n*2+1]
__global__ void logits_final_kernel(const float* __restrict__ H2,
                                    const float* __restrict__ W3,
                                    const float* __restrict__ b3,
                                    float* __restrict__ out, int M) {
  int n = blockIdx.x * blockDim.x + threadIdx.x;
  if (n >= M) return;
  float a0 = b3[0], a1 = b3[1];
  const float* h = H2 + (size_t)n * 64;
#pragma unroll
  for (int k = 0; k < 64; ++k) {
    float v = h[k];
    a0 += v * W3[k * 2 + 0];
    a1 += v * W3[k * 2 + 1];
  }
  out[(size_t)n * 2 + 0] = a0;
  out[(size_t)n * 2 + 1] = a1;
}

__global__ void zero_kernel(float* __restrict__ p, int n) {
  int i = blockIdx.x * blockDim.x + threadIdx.x;
  if (i < n) p[i] = 0.0f;
}

// partial column sums over 512 nodes per block, accumulated atomically
__global__ __launch_bounds__(128) void pool_kernel(const float* __restrict__ X,
                                                   float* __restrict__ pool) {
  const int ccol = threadIdx.x;
  const size_t n0 = (size_t)blockIdx.x * 512;
  float s = 0.0f;
  for (int i = 0; i < 512; ++i) s += X[(n0 + i) * HID + ccol];
  atomicAdd(&pool[ccol], s);
}

// single-block value MLP: 128 -> 128 -> 64 -> 1
__global__ __launch_bounds__(128) void value_mlp_kernel(
    const float* __restrict__ pool, const float* __restrict__ W1,
    const float* __restrict__ b1, const float* __restrict__ W2,
    const float* __restrict__ b2, const float* __restrict__ W3,
    const float* __restrict__ b3, float* __restrict__ out, float invN) {
  __shared__ float p[128], h1[128], h2[64];
  const int t = threadIdx.x;
  p[t] = pool[t] * invN;
  __syncthreads();
  float a = b1[t];
  for (int k = 0; k < 128; ++k) a += p[k] * W1[k * 128 + t];
  h1[t] = fmaxf(a, 0.0f);
  __syncthreads();
  if (t < 64) {
    float a2 = b2[t];
    for (int k = 0; k < 128; ++k) a2 += h1[k] * W2[k * 64 + t];
    h2[t] = fmaxf(a2, 0.0f);
  }
  __syncthreads();
  if (t == 0) {
    float a3 = b3[0];
    for (int k = 0; k < 64; ++k) a3 += h2[k] * W3[k];
    out[0] = a3;
  }
}

// ---------------------------------------------------------------------------
extern "C" void kernel_launch(void* const* d_in, const int* in_sizes, int n_in,
                              void* d_out, int out_size, void* d_ws,
                              size_t ws_size, hipStream_t stream) {
  const float* nf = (const float*)d_in[0];
  const int* ei = (const int*)d_in[1];
  const float* W_emb = (const float*)d_in[2];
  const float* b_emb = (const float*)d_in[3];
  const float* ln0_g = (const float*)d_in[4];
  const float* ln0_b = (const float*)d_in[5];
  const float* Wg = (const float*)d_in[6];
  const float* att_s = (const float*)d_in[7];
  const float* att_d = (const float*)d_in[8];
  const float* bg = (const float*)d_in[9];
  const float* lng = (const float*)d_in[10];
  const float* lnb = (const float*)d_in[11];
  const float* pW1 = (const float*)d_in[12];
  const float* pb1 = (const float*)d_in[13];
  const float* pW2 = (const float*)d_in[14];
  const float* pb2 = (const float*)d_in[15];
  const float* pW3 = (const float*)d_in[16];
  const float* pb3 = (const float*)d_in[17];
  const float* vW1 = (const float*)d_in[18];
  const float* vb1 = (const float*)d_in[19];
  const float* vW2 = (const float*)d_in[20];
  const float* vb2 = (const float*)d_in[21];
  const float* vW3 = (const float*)d_in[22];
  const float* vb3 = (const float*)d_in[23];

  const int N = in_sizes[0] / F_IN;        // 65536
  const int E = in_sizes[1] / 2;           // 524288
  const int NDATA = (out_size - 1) / 2;    // 32768
  const int* src = ei;
  const int* dst = ei + E;

  // workspace layout (floats)
  float* ws = (float*)d_ws;
  const size_t NH = (size_t)N * HID;
  const size_t N4 = (size_t)N * HEADS;
  float* X = ws;
  float* Hb = X + NH;
  float* OUT = Hb + NH;
  float* AS = OUT + NH;
  float* AD = AS + N4;
  float* MX = AD + N4;
  float* DEN = MX + N4;
  float* WEDGE = DEN + N4;               // E*HEADS
  float* POOL = WEDGE + (size_t)E * HEADS;

  const int NH4 = N * HEADS;
  const int EH4 = E * HEADS;

  // 1) node embedding + LN + relu
  embed_ln_kernel<<<N, 128, 0, stream>>>(nf, W_emb, b_emb, ln0_g, ln0_b, X);

  // 2) GAT layers
  for (int l = 0; l < LAYERS; ++l) {
    const float* Wl = Wg + (size_t)l * HID * HID;
    // h = x @ Wg[l]   (WMMA fp32)
    int tiles = (N / 16) * (HID / 16);
    gemm_wmma_f32<<<tiles / 8, 256, 0, stream>>>(X, Wl, nullptr, Hb, N, HID,
                                                 HID, 0);
    attn_dots_kernel<<<N, 128, 0, stream>>>(Hb, att_s + (size_t)l * HID,
                                            att_d + (size_t)l * HID, AS, AD);
    init_max_kernel<<<(NH4 + 255) / 256, 256, 0, stream>>>(AS, AD, MX, NH4);
    edge_max_kernel<<<(EH4 + 255) / 256, 256, 0, stream>>>(src, dst, AS, AD,
                                                           MX, EH4);
    init_den_kernel<<<(NH4 + 255) / 256, 256, 0, stream>>>(AS, AD, MX, DEN,
                                                           NH4);
    edge_expsum_kernel<<<(EH4 + 255) / 256, 256, 0, stream>>>(
        src, dst, AS, AD, MX, DEN, WEDGE, EH4);
    agg_init_kernel<<<N, 128, 0, stream>>>(Hb, AS, AD, MX, DEN, OUT);
    edge_agg_kernel<<<(E + 1) / 2, 256, 0, stream>>>(src, dst, Hb, WEDGE, DEN,
                                                     OUT, E);
    post_ln_kernel<<<N, 128, 0, stream>>>(X, OUT, bg + (size_t)l * HID,
                                          lng + (size_t)l * HID,
                                          lnb + (size_t)l * HID);
  }

  // 3) policy head on first NDATA nodes (WMMA fp32 GEMMs + tiny final layer)
  {
    int tiles1 = (NDATA / 16) * (HID / 16);
    gemm_wmma_f32<<<tiles1 / 8, 256, 0, stream>>>(X, pW1, pb1, Hb, NDATA, HID,
                                                  HID, 1);
    int tiles2 = (NDATA / 16) * (64 / 16);
    gemm_wmma_f32<<<tiles2 / 8, 256, 0, stream>>>(Hb, pW2, pb2, OUT, NDATA,
                                                  HID, 64, 1);
    logits_final_kernel<<<(NDATA + 255) / 256, 256, 0, stream>>>(
        OUT, pW3, pb3, (float*)d_out, NDATA);
  }

  // 4) value head: mean pool over all nodes, then 128->128->64->1
  zero_kernel<<<1, 128, 0, stream>>>(POOL, HID);
  pool_kernel<<<N / 512, 128, 0, stream>>>(X, POOL);
  value_mlp_kernel<<<1, 128, 0, stream>>>(POOL, vW1, vb1, vW2, vb2, vW3, vb3,
                                          (float*)d_out + (size_t)NDATA * 2,
                                          1.0f / (float)N);
}